// CausalInterventionAttention_64244120814029
// MI455X (gfx1250) — compile-verified
//
#include <hip/hip_runtime.h>
#include <hip/hip_bf16.h>

// CDNA5 / gfx1250, wave32. fp32 matrix path via V_WMMA_F32_16X16X4_F32.
typedef __attribute__((ext_vector_type(2))) float v2f;
typedef __attribute__((ext_vector_type(8))) float v8f;

// Only check builtin availability in the DEVICE pass; the host (x86) pass
// reports 0 for all amdgcn builtins.
#if defined(__HIP_DEVICE_COMPILE__) && \
    !__has_builtin(__builtin_amdgcn_wmma_f32_16x16x4_f32)
#error "gfx1250 wmma f32 16x16x4 builtin not available"
#endif

static __device__ __forceinline__ v8f wmma4(v2f a, v2f b, v8f c) {
  // 8 args: (neg_a, A, neg_b, B, c_mod, C, reuse_a, reuse_b)
  return __builtin_amdgcn_wmma_f32_16x16x4_f32(false, a, false, b, (short)0, c,
                                               false, false);
}

#define BB   2
#define SS   2048
#define DD   1024
#define HH   16
#define HD   64
#define MR   (BB * SS)   // 4096 rows of the token matrix

// ---------------------------------------------------------------------------
// Kernel 1: fused Q/K/V projection.  x(4096x1024) @ W(1024x1024) + b.
// One wave -> 32x64 output tile: 2 A-frags (rows m0 / m0+16) share 4 B-frags
// per K-step, halving B-load traffic per WMMA.
// Result scattered to (B,H,S,HD) layout for the attention stage.
// ---------------------------------------------------------------------------
__global__ __launch_bounds__(256) void qkv_proj(
    const float* __restrict__ x,
    const float* __restrict__ Wq, const float* __restrict__ bq,
    const float* __restrict__ Wk, const float* __restrict__ bk,
    const float* __restrict__ Wv, const float* __restrict__ bv,
    float* __restrict__ qo, float* __restrict__ ko, float* __restrict__ vo) {
  const int z = blockIdx.z;
  const float* __restrict__ W    = (z == 0) ? Wq : ((z == 1) ? Wk : Wv);
  const float* __restrict__ bias = (z == 0) ? bq : ((z == 1) ? bk : bv);
  float* __restrict__ dst        = (z == 0) ? qo : ((z == 1) ? ko : vo);

  const int wave = threadIdx.x >> 5;
  const int lane = threadIdx.x & 31;
  const int ll = lane & 15;     // N index within 16-wide frag / A row
  const int lh = lane >> 4;     // half selector (K offset 0 or 2)

  const int tile  = blockIdx.x * 8 + wave;  // 0..2047
  const int mtile = tile >> 4;              // 0..127 (32-row tiles)
  const int ngrp  = tile & 15;              // 0..15
  const int m0 = mtile * 32;
  const int n0 = ngrp * 64;

  v8f acc[2][4];
#pragma unroll
  for (int nf = 0; nf < 4; ++nf) {
    const float bvv = bias[n0 + nf * 16 + ll];
#pragma unroll
    for (int r = 0; r < 8; ++r) { acc[0][nf][r] = bvv; acc[1][nf][r] = bvv; }
  }

  const float* __restrict__ xrow0 = x + (size_t)(m0 + ll) * DD + lh * 2;
  const float* __restrict__ xrow1 = xrow0 + (size_t)16 * DD;
  for (int k0 = 0; k0 < DD; k0 += 4) {
    const v2f a0 = *(const v2f*)(xrow0 + k0);
    const v2f a1 = *(const v2f*)(xrow1 + k0);
    const float* __restrict__ wr = W + (size_t)(k0 + lh * 2) * DD + n0 + ll;
#pragma unroll
    for (int nf = 0; nf < 4; ++nf) {
      v2f bf;
      bf.x = wr[nf * 16];        // K = k0 + lh*2
      bf.y = wr[DD + nf * 16];   // K = k0 + lh*2 + 1
      acc[0][nf] = wmma4(a0, bf, acc[0][nf]);
      acc[1][nf] = wmma4(a1, bf, acc[1][nf]);
    }
  }

  // Scatter (m,n) -> (b, h, s, hd)
#pragma unroll
  for (int mt = 0; mt < 2; ++mt) {
#pragma unroll
    for (int nf = 0; nf < 4; ++nf) {
      const int n = n0 + nf * 16 + ll;
      const int h = n >> 6, hd = n & 63;
#pragma unroll
      for (int r = 0; r < 8; ++r) {
        const int m = m0 + mt * 16 + r + lh * 8;
        const int b = m >> 11, s = m & 2047;
        dst[(((size_t)b * HH + h) * SS + s) * HD + hd] = acc[mt][nf][r];
      }
    }
  }
}

// ---------------------------------------------------------------------------
// Kernel 2: flash-style attention with intervention mask.
// One wave owns 16 query rows; streams 16-key tiles; online softmax with
// 16-lane butterfly shuffles; P is transposed C->A layout via wave-private LDS.
// ---------------------------------------------------------------------------
__global__ __launch_bounds__(256) void attn_kernel(
    const float* __restrict__ qws, const float* __restrict__ kws,
    const float* __restrict__ vws,
    const unsigned char* __restrict__ cause,
    const unsigned char* __restrict__ effect,
    const float* __restrict__ strength_p,
    float* __restrict__ outws /* (B,S,D) */) {
  __shared__ float plds[8 * 256];  // 16x16 P tile per wave

  const int wave = threadIdx.x >> 5;
  const int lane = threadIdx.x & 31;
  const int ll = lane & 15, lh = lane >> 4;

  const int bh   = blockIdx.x >> 4;   // 0..31
  const int qblk = blockIdx.x & 15;   // 0..15
  const int b = bh >> 4, h = bh & 15;
  const int q0 = qblk * 128 + wave * 16;

  const float strength = strength_p[0];
  const float fm = 1.0f - 0.5f * strength;  // factor where cause & effect

  const float* __restrict__ qp = qws + ((size_t)bh * SS + q0) * HD;
  const float* __restrict__ kp = kws + (size_t)bh * SS * HD;
  const float* __restrict__ vp = vws + (size_t)bh * SS * HD;

  // Q A-frags: (M=ll, K=i*4 + lh*2 .. +1), kept resident for the whole loop.
  v2f qa[16];
#pragma unroll
  for (int i = 0; i < 16; ++i)
    qa[i] = *(const v2f*)(qp + ll * HD + i * 4 + lh * 2);

  // Per-lane cause bits for the rows this lane owns in the C/D layout.
  bool cb[8];
#pragma unroll
  for (int r = 0; r < 8; ++r)
    cb[r] = cause[b * SS + q0 + r + lh * 8] != 0;

  float mrow[8], lrow[8];
  v8f o[4];
#pragma unroll
  for (int r = 0; r < 8; ++r) { mrow[r] = -__builtin_inff(); lrow[r] = 0.0f; }
#pragma unroll
  for (int nf = 0; nf < 4; ++nf)
#pragma unroll
    for (int r = 0; r < 8; ++r) o[nf][r] = 0.0f;

  float* __restrict__ pl = plds + wave * 256;

  for (int k0 = 0; k0 < SS; k0 += 16) {
    const bool eff = effect[b * SS + k0 + ll] != 0;  // N = ll for this lane

    // scores S = Q @ K^T  (16 K-steps of 4 over HD=64)
    v8f s;
#pragma unroll
    for (int r = 0; r < 8; ++r) s[r] = 0.0f;
#pragma unroll
    for (int i = 0; i < 16; ++i) {
      const v2f kb = *(const v2f*)(kp + (size_t)(k0 + ll) * HD + i * 4 + lh * 2);
      s = wmma4(qa[i], kb, s);
    }

    // scale 1/sqrt(64), intervention factor, online softmax
    float p[8], alpha[8];
#pragma unroll
    for (int r = 0; r < 8; ++r) {
      const float f = (cb[r] && eff) ? fm : 1.0f;
      p[r] = s[r] * 0.125f * f;
    }
#pragma unroll
    for (int r = 0; r < 8; ++r) {
      float v = p[r];
      v = fmaxf(v, __shfl_xor(v, 1));
      v = fmaxf(v, __shfl_xor(v, 2));
      v = fmaxf(v, __shfl_xor(v, 4));
      v = fmaxf(v, __shfl_xor(v, 8));
      const float mnew = fmaxf(mrow[r], v);
      alpha[r] = __expf(mrow[r] - mnew);
      mrow[r] = mnew;
      p[r] = __expf(p[r] - mnew);
      float sum = p[r];
      sum += __shfl_xor(sum, 1);
      sum += __shfl_xor(sum, 2);
      sum += __shfl_xor(sum, 4);
      sum += __shfl_xor(sum, 8);
      lrow[r] = lrow[r] * alpha[r] + sum;
    }
#pragma unroll
    for (int nf = 0; nf < 4; ++nf)
#pragma unroll
      for (int r = 0; r < 8; ++r) o[nf][r] *= alpha[r];

    // C-layout -> A-layout transpose of P through wave-private LDS
#pragma unroll
    for (int r = 0; r < 8; ++r)
      pl[(r + lh * 8) * 16 + ll] = p[r];
    asm volatile("s_wait_dscnt 0" ::: "memory");  // cross-lane LDS dependency
    v2f pa[4];
#pragma unroll
    for (int i = 0; i < 4; ++i)
      pa[i] = *(const v2f*)(pl + ll * 16 + i * 4 + lh * 2);

    // O += P @ V  (K=16 keys in 4 steps, N=64 in 4 frags)
#pragma unroll
    for (int nf = 0; nf < 4; ++nf) {
#pragma unroll
      for (int i = 0; i < 4; ++i) {
        const float* __restrict__ vb0 =
            vp + (size_t)(k0 + i * 4 + lh * 2) * HD + nf * 16 + ll;
        v2f vb;
        vb.x = vb0[0];
        vb.y = vb0[HD];
        o[nf] = wmma4(pa[i], vb, o[nf]);
      }
    }
  }

  // normalize, write back to (B,S,D)
#pragma unroll
  for (int nf = 0; nf < 4; ++nf) {
#pragma unroll
    for (int r = 0; r < 8; ++r) {
      const int qrow = q0 + r + lh * 8;
      outws[(size_t)(b * SS + qrow) * DD + h * HD + nf * 16 + ll] =
          o[nf][r] / lrow[r];
    }
  }
}

// ---------------------------------------------------------------------------
// Kernel 3: output projection.  attn(4096x1024) @ Wo + bo -> d_out row-major.
// Same 32x64-per-wave tiling as qkv_proj.
// ---------------------------------------------------------------------------
__global__ __launch_bounds__(256) void out_proj(
    const float* __restrict__ A, const float* __restrict__ W,
    const float* __restrict__ bias, float* __restrict__ out) {
  const int wave = threadIdx.x >> 5;
  const int lane = threadIdx.x & 31;
  const int ll = lane & 15, lh = lane >> 4;

  const int tile  = blockIdx.x * 8 + wave;  // 0..2047
  const int mtile = tile >> 4;
  const int ngrp  = tile & 15;
  const int m0 = mtile * 32;
  const int n0 = ngrp * 64;

  v8f acc[2][4];
#pragma unroll
  for (int nf = 0; nf < 4; ++nf) {
    const float bvv = bias[n0 + nf * 16 + ll];
#pragma unroll
    for (int r = 0; r < 8; ++r) { acc[0][nf][r] = bvv; acc[1][nf][r] = bvv; }
  }

  const float* __restrict__ arow0 = A + (size_t)(m0 + ll) * DD + lh * 2;
  const float* __restrict__ arow1 = arow0 + (size_t)16 * DD;
  for (int k0 = 0; k0 < DD; k0 += 4) {
    const v2f a0 = *(const v2f*)(arow0 + k0);
    const v2f a1 = *(const v2f*)(arow1 + k0);
    const float* __restrict__ wr = W + (size_t)(k0 + lh * 2) * DD + n0 + ll;
#pragma unroll
    for (int nf = 0; nf < 4; ++nf) {
      v2f bf;
      bf.x = wr[nf * 16];
      bf.y = wr[DD + nf * 16];
      acc[0][nf] = wmma4(a0, bf, acc[0][nf]);
      acc[1][nf] = wmma4(a1, bf, acc[1][nf]);
    }
  }

#pragma unroll
  for (int mt = 0; mt < 2; ++mt) {
#pragma unroll
    for (int nf = 0; nf < 4; ++nf) {
      const int n = n0 + nf * 16 + ll;
#pragma unroll
      for (int r = 0; r < 8; ++r) {
        const int m = m0 + mt * 16 + r + lh * 8;
        out[(size_t)m * DD + n] = acc[mt][nf][r];
      }
    }
  }
}

// ---------------------------------------------------------------------------
extern "C" void kernel_launch(void* const* d_in, const int* in_sizes, int n_in,
                              void* d_out, int out_size, void* d_ws,
                              size_t ws_size, hipStream_t stream) {
  (void)in_sizes; (void)n_in; (void)out_size; (void)ws_size;

  const float* x = (const float*)d_in[0];
  const unsigned char* cause  = (const unsigned char*)d_in[1];  // bool (1B)
  const unsigned char* effect = (const unsigned char*)d_in[2];  // bool (1B)
  const float* strength = (const float*)d_in[3];
  const float* Wq = (const float*)d_in[4];
  const float* bq = (const float*)d_in[5];
  const float* Wk = (const float*)d_in[6];
  const float* bk = (const float*)d_in[7];
  const float* Wv = (const float*)d_in[8];
  const float* bv = (const float*)d_in[9];
  const float* Wo = (const float*)d_in[10];
  const float* bo = (const float*)d_in[11];
  float* out = (float*)d_out;

  float* ws = (float*)d_ws;
  const size_t per = (size_t)BB * SS * DD;  // 4,194,304 floats
  float* qws = ws;
  float* kws = ws + per;
  float* vws = ws + 2 * per;
  float* aws = ws + 3 * per;  // total 64 MB of scratch

  dim3 blk(256, 1, 1);
  qkv_proj<<<dim3(256, 1, 3), blk, 0, stream>>>(x, Wq, bq, Wk, bk, Wv, bv,
                                                qws, kws, vws);
  attn_kernel<<<dim3(512, 1, 1), blk, 0, stream>>>(qws, kws, vws, cause,
                                                   effect, strength, aws);
  out_proj<<<dim3(256, 1, 1), blk, 0, stream>>>(aws, Wo, bo, out);
}